// RegionLoss2_72473278153176
// MI455X (gfx1250) — compile-verified
//
#include <hip/hip_runtime.h>
#include <hip/hip_bf16.h>
#include <math.h>

// ---------------- problem constants ----------------
#define NBATCH 64
#define NANCH  5
#define NCLS   20
#define NHH    52
#define NWW    52
#define GMAX   50
#define NHW    (NHH * NWW)          // 2704
#define CELLS_PER_B (NANCH * NHW)   // 13520
#define BLK    256
#define NBLK_Y ((CELLS_PER_B + BLK - 1) / BLK)  // 53
#define NPART  (NBATCH * NBLK_Y)                // 3392 (divisible by 32)

#define OBJECT_SCALE 5.0f
#define THRESH       0.6f

__constant__ float c_aw[NANCH] = {1.08f, 3.42f, 6.63f, 9.42f, 16.62f};
__constant__ float c_ah[NANCH] = {1.19f, 4.41f, 11.38f, 5.11f, 10.52f};

// ---------------- CDNA5 feature detection (device pass only) ----------------
#if defined(__HIP_DEVICE_COMPILE__) && defined(__gfx1250__)
  #if __has_builtin(__builtin_amdgcn_wmma_f32_16x16x4_f32)
    #define USE_WMMA 1
  #endif
  #if __has_builtin(__builtin_amdgcn_global_load_async_to_lds_b128) && \
      __has_builtin(__builtin_amdgcn_global_load_async_to_lds_b32)
    #define USE_ASYNC 1
  #endif
  #if __has_builtin(__builtin_amdgcn_s_wait_asynccnt)
    #define WAIT_ASYNC() __builtin_amdgcn_s_wait_asynccnt(0)
  #else
    #define WAIT_ASYNC() asm volatile("s_wait_asynccnt 0" ::: "memory")
  #endif
#endif

typedef float v2f __attribute__((ext_vector_type(2)));
typedef float v8f __attribute__((ext_vector_type(8)));
typedef int   v4i __attribute__((ext_vector_type(4)));

// IoU on center-format boxes, matching reference iou_centers()
__device__ __forceinline__ float iou_centers(float x1, float y1, float w1, float h1,
                                             float x2, float y2, float w2, float h2) {
  float uw = fmaxf(x1 + w1 * 0.5f, x2 + w2 * 0.5f) - fminf(x1 - w1 * 0.5f, x2 - w2 * 0.5f);
  float uh = fmaxf(y1 + h1 * 0.5f, y2 + h2 * 0.5f) - fminf(y1 - h1 * 0.5f, y2 - h2 * 0.5f);
  float cw = w1 + w2 - uw;
  float ch = h1 + h2 - uh;
  float inter = (cw <= 0.0f || ch <= 0.0f) ? 0.0f : cw * ch;
  return inter / (w1 * h1 + w2 * h2 - inter);
}

__device__ __forceinline__ float sigmoidf(float v) { return 1.0f / (1.0f + expf(-v)); }

// ---------------- kernel 1: per-GT preprocessing (sequential validity per batch) ----
__global__ void k_prep(const float* __restrict__ target,
                       float4* __restrict__ gtbox, float4* __restrict__ gtt,
                       float* __restrict__ gtcls, int* __restrict__ gtcell) {
  int b = blockIdx.x * blockDim.x + threadIdx.x;
  if (b >= NBATCH) return;
  bool vflag = true;
  for (int g = 0; g < GMAX; ++g) {
    const float* t = target + ((size_t)b * GMAX + g) * 5;
    float cls = t[0], xs = t[1], ys = t[2], ws = t[3], hs = t[4];
    vflag = vflag && (xs != 0.0f);              // cumprod(x != 0) > 0
    float gx = xs * (float)NWW, gy = ys * (float)NHH;
    float gw = ws * (float)NWW, gh = hs * (float)NHH;
    // best anchor: IoU of (0,0,gw,gh) vs (0,0,aw,ah) -> min/min over max/max
    int bestn = 0; float best = -1.0f;
    #pragma unroll
    for (int a = 0; a < NANCH; ++a) {
      float aw = c_aw[a], ah = c_ah[a];
      float inter = fminf(gw, aw) * fminf(gh, ah);
      float iou = inter / (gw * gh + aw * ah - inter);
      if (iou > best) { best = iou; bestn = a; }   // strict '>' == argmax-first
    }
    int gi = (int)gx, gj = (int)gy;
    float baw = c_aw[bestn], bah = c_ah[bestn];
    int idx = b * GMAX + g;
    gtbox[idx] = make_float4(gx, gy, gw, gh);
    gtt[idx]   = make_float4(gx - (float)gi, gy - (float)gj,
                             logf(gw / baw), logf(gh / bah));
    gtcls[idx] = cls;
    gtcell[idx] = vflag ? ((bestn * NHH + gj) * NWW + gi) : -1;
  }
}

// ---------------- kernel 2: main per-cell pass + block reduction --------------------
__global__ __launch_bounds__(BLK) void k_main(
    const float* __restrict__ out, const int* __restrict__ seenp,
    const float4* __restrict__ gtbox, const float4* __restrict__ gtt,
    const float* __restrict__ gtcls, const int* __restrict__ gtcell,
    float* __restrict__ partials)
{
  __shared__ float4 sbox[GMAX];
  __shared__ float4 stt[GMAX];
  __shared__ float  scls[GMAX];
  __shared__ int    scell[GMAX];
  __shared__ float  red[BLK];

  const int b   = blockIdx.x;
  const int tid = threadIdx.x;

  // ---- stage per-batch GT metadata into LDS (CDNA5 async-to-LDS when available) ----
#if defined(USE_ASYNC)
  if (tid < GMAX) {
    __builtin_amdgcn_global_load_async_to_lds_b128((v4i*)&gtbox[b * GMAX + tid],
                                                   (v4i*)&sbox[tid], 0, 0);
  } else if (tid < 2 * GMAX) {
    int g = tid - GMAX;
    __builtin_amdgcn_global_load_async_to_lds_b128((v4i*)&gtt[b * GMAX + g],
                                                   (v4i*)&stt[g], 0, 0);
  } else if (tid < 3 * GMAX) {
    int g = tid - 2 * GMAX;
    __builtin_amdgcn_global_load_async_to_lds_b32((int*)&gtcls[b * GMAX + g],
                                                  (int*)&scls[g], 0, 0);
  } else if (tid < 4 * GMAX) {
    int g = tid - 3 * GMAX;
    __builtin_amdgcn_global_load_async_to_lds_b32((int*)&gtcell[b * GMAX + g],
                                                  (int*)&scell[g], 0, 0);
  }
  WAIT_ASYNC();
#else
  if (tid < GMAX) {
    sbox[tid]  = gtbox[b * GMAX + tid];
    stt[tid]   = gtt[b * GMAX + tid];
    scls[tid]  = gtcls[b * GMAX + tid];
    scell[tid] = gtcell[b * GMAX + tid];
  }
#endif
  __syncthreads();

  float loss = 0.0f;
  const bool early = (seenp[0] < 12800);
  const int cell = blockIdx.y * BLK + tid;
  if (cell < CELLS_PER_B) {
    const int a   = cell / NHW;
    const int rem = cell - a * NHW;
    const int j   = rem / NWW;
    const int i   = rem - j * NWW;
    const float* base =
        out + ((size_t)b * (NANCH * (5 + NCLS)) + (size_t)a * (5 + NCLS)) * NHW + rem;

    const float o0 = base[0 * NHW], o1 = base[1 * NHW];
    const float o2 = base[2 * NHW], o3 = base[3 * NHW], o4 = base[4 * NHW];
    const float xs = sigmoidf(o0), ys = sigmoidf(o1), conf = sigmoidf(o4);
    const float px = xs + (float)i, py = ys + (float)j;
    const float pw = expf(o2) * c_aw[a], ph = expf(o3) * c_ah[a];

    // max IoU vs all valid GT boxes + resolve scatter assignment (largest g wins)
    float cur = 0.0f;
    int ag = -1;
    for (int g = 0; g < GMAX; ++g) {
      const int gc = scell[g];
      if (gc < 0) continue;
      const float4 gb = sbox[g];
      const float iou = iou_centers(px, py, pw, ph, gb.x, gb.y, gb.z, gb.w);
      cur = fmaxf(cur, iou);
      if (gc == cell) ag = g;
    }

    if (ag >= 0) {
      // assigned cell: coord targets from GT, conf target = IoU(gt, pred), CE class loss
      const float4 tv = stt[ag];
      const float dx = xs - tv.x, dy = ys - tv.y, dw = o2 - tv.z, dh = o3 - tv.w;
      loss += 0.5f * (dx * dx + dy * dy + dw * dw + dh * dh);

      const float4 gb = sbox[ag];
      const float ti = iou_centers(px, py, pw, ph, gb.x, gb.y, gb.z, gb.w);
      const float dc = conf - ti;
      loss += 0.5f * OBJECT_SCALE * dc * dc;

      int c = (int)scls[ag];
      c = (c < 0) ? 0 : ((c > NCLS - 1) ? NCLS - 1 : c);
      float m = -3.4e38f, lsel = 0.0f;
      #pragma unroll
      for (int k = 0; k < NCLS; ++k) {
        float v = base[(5 + k) * NHW];
        m = fmaxf(m, v);
        lsel = (k == c) ? v : lsel;
      }
      float s = 0.0f;
      #pragma unroll
      for (int k = 0; k < NCLS; ++k) s += expf(base[(5 + k) * NHW] - m);
      loss += (m + logf(s)) - lsel;      // -log_softmax[c]
    } else {
      if (early) {                        // seen < 12800: coord_mask=1, tx=ty=0.5
        const float dx = xs - 0.5f, dy = ys - 0.5f;
        loss += 0.5f * (dx * dx + dy * dy + o2 * o2 + o3 * o3);
      }
      const float cmask = (cur > THRESH) ? 0.0f : 1.0f;  // NOOBJECT_SCALE = 1
      loss += 0.5f * cmask * conf * conf;
    }
  }

  red[tid] = loss;
  __syncthreads();

#if defined(USE_WMMA)
  // Wave 0 reduces all 256 block partials with 4 chained V_WMMA_F32_16X16X4_F32:
  // A = ones(16x4) -> D[m][n] = sum_k B[k][n] + C[m][n]; every B slot holds one partial,
  // so row 0 of the final D holds 16 column sums whose total is the block sum (all f32).
  if (tid < 32) {
    v8f acc = {};
    const v2f ones = {1.0f, 1.0f};
    #pragma unroll
    for (int jx = 0; jx < BLK / 64; ++jx) {
      v2f bv;
      bv.x = red[jx * 64 + tid];
      bv.y = red[jx * 64 + 32 + tid];
      acc = __builtin_amdgcn_wmma_f32_16x16x4_f32(false, ones, false, bv,
                                                  (short)0, acc, false, false);
    }
    red[tid] = acc[0];   // lanes 0..15 of VGPR0 = D row 0 (16 column sums)
  }
  __syncthreads();
  if (tid == 0) {
    float s = 0.0f;
    #pragma unroll
    for (int k = 0; k < 16; ++k) s += red[k];
    partials[blockIdx.x * NBLK_Y + blockIdx.y] = s;
  }
#else
  for (int off = BLK / 2; off > 0; off >>= 1) {
    if (tid < off) red[tid] += red[tid + off];
    __syncthreads();
  }
  if (tid == 0) partials[blockIdx.x * NBLK_Y + blockIdx.y] = red[0];
#endif
}

// ---------------- kernel 3: deterministic final reduction (single wave) -------------
__global__ void k_final(const float* __restrict__ partials, float* __restrict__ outv) {
  __shared__ float red[32];
  const int lane = threadIdx.x;
  float p = 0.0f;
  for (int i = lane; i < NPART; i += 32) p += partials[i];   // NPART % 32 == 0: uniform
#if defined(USE_WMMA)
  const v2f ones = {1.0f, 1.0f};
  v2f bv = {p, 0.0f};
  v8f acc = {};
  acc = __builtin_amdgcn_wmma_f32_16x16x4_f32(false, ones, false, bv,
                                              (short)0, acc, false, false);
  red[lane] = acc[0];
  __syncthreads();
  if (lane == 0) {
    float s = 0.0f;
    #pragma unroll
    for (int k = 0; k < 16; ++k) s += red[k];
    outv[0] = s;
  }
#else
  red[lane] = p;
  __syncthreads();
  if (lane == 0) {
    float s = 0.0f;
    for (int k = 0; k < 32; ++k) s += red[k];
    outv[0] = s;
  }
#endif
}

// ---------------- host launcher -----------------------------------------------------
extern "C" void kernel_launch(void* const* d_in, const int* in_sizes, int n_in,
                              void* d_out, int out_size, void* d_ws, size_t ws_size,
                              hipStream_t stream) {
  (void)in_sizes; (void)n_in; (void)out_size; (void)ws_size;

  const float* output = (const float*)d_in[0];
  const int*   seen   = (const int*)d_in[1];
  const float* target = (const float*)d_in[2];
  float* outv = (float*)d_out;

  char* ws = (char*)d_ws;
  float4* gtbox = (float4*)ws;  ws += sizeof(float4) * NBATCH * GMAX;
  float4* gtt   = (float4*)ws;  ws += sizeof(float4) * NBATCH * GMAX;
  float*  gtcls = (float*)ws;   ws += sizeof(float)  * NBATCH * GMAX;
  int*    gtcell= (int*)ws;     ws += sizeof(int)    * NBATCH * GMAX;
  float*  partials = (float*)ws;  // NPART floats

  k_prep<<<1, 64, 0, stream>>>(target, gtbox, gtt, gtcls, gtcell);
  dim3 grid(NBATCH, NBLK_Y);
  k_main<<<grid, BLK, 0, stream>>>(output, seen, gtbox, gtt, gtcls, gtcell, partials);
  k_final<<<1, 32, 0, stream>>>(partials, outv);
}